// IntervalLoss_86242943304042
// MI455X (gfx1250) — compile-verified
//
#include <hip/hip_runtime.h>

// ---------------------------------------------------------------------------
// IntervalLoss for MI455X (gfx1250, wave32).
// Streaming, HBM-bound: 268MB read @ 23.3 TB/s => ~11.5us floor.
// Non-temporal B128 loads (single-use data > 192MB L2), branchless interval
// math, WMMA-based wave-32 reduction (V_WMMA_F32_16X16X4_F32 with B=ones).
// ---------------------------------------------------------------------------

typedef float vf4 __attribute__((ext_vector_type(4)));
typedef float v2f __attribute__((ext_vector_type(2)));
typedef float v8f __attribute__((ext_vector_type(8)));

// Full wave32 sum using one WMMA.
// A (16x4 f32, 2 VGPRs/lane): lane L in [0,16): VGPR0=A[L][0], VGPR1=A[L][1];
// lane L in [16,32): VGPR0=A[L-16][2], VGPR1=A[L-16][3].
// Put partial in VGPR0, 0 in VGPR1; B = all ones (layout-independent).
// D[m][n] = sum_k A[m][k] = partial_m + partial_{m+16}, identical for all n.
// D layout: lane L<16 VGPR j = D[j][L]; lane L>=16 VGPR j = D[j+8][L-16].
// Sum of 8 D VGPRs: lanes 0..15 hold rows 0..7 sums, lanes 16..31 rows 8..15.
// One shfl_xor(16) combines them -> total wave sum in every lane.
__device__ __forceinline__ float wave_sum_wmma(float x) {
    v2f a; a[0] = x;    a[1] = 0.0f;
    v2f b; b[0] = 1.0f; b[1] = 1.0f;
    v8f c = {};
    v8f d = __builtin_amdgcn_wmma_f32_16x16x4_f32(
        /*neg_a=*/false, a, /*neg_b=*/false, b,
        /*c_mod=*/(short)0, c, /*reuse_a=*/false, /*reuse_b=*/false);
    float s = d[0] + d[1] + d[2] + d[3] + d[4] + d[5] + d[6] + d[7];
    s += __shfl_xor(s, 16, 32);
    return s;
}

__device__ __forceinline__ float elem_loss(float p, float t) {
    float d = p - t;
    float loss = d * d;  // default MSE (covers mask_zero + remaining branches)
    constexpr float C[11]  = {0.05f, 0.125f, 0.225f, 0.4f, 0.5f, 0.6f,
                              0.75f, 0.95f, 1.6f, 2.0f, 2.5f};
    constexpr float LO[11] = {0.0f, 0.0f, 0.15f, 0.3f, 0.3f, 0.3f,
                              0.7f, 0.7f, 1.2f, 1.2f, 1.2f};
    constexpr float HI[11] = {0.1f, 0.15f, 0.3f, 0.7f, 0.7f, 0.7f,
                              1.2f, 1.2f, 2.5f, 3.0f, 5.0f};
#pragma unroll
    for (int i = 0; i < 11; ++i) {
        // interval penalty: (lo-p)^2 if p<lo, (p-hi)^2 if p>hi, else 0
        // == max(lo-p, p-hi, 0)^2  (lo<hi so at most one positive) -> v_max3
        float e = fmaxf(fmaxf(LO[i] - p, p - HI[i]), 0.0f);
        loss = (fabsf(t - C[i]) < 0.01f) ? (e * e) : loss;  // v_cmp + cndmask
    }
    return loss;
}

__global__ void __launch_bounds__(256)
interval_loss_partial(const float* __restrict__ pred,
                      const float* __restrict__ tgt,
                      float* __restrict__ partials, long long n) {
    const long long n4     = n >> 2;
    const long long tid    = (long long)blockIdx.x * blockDim.x + threadIdx.x;
    const long long stride = (long long)gridDim.x * blockDim.x;
    const vf4* __restrict__ p4 = (const vf4*)pred;
    const vf4* __restrict__ t4 = (const vf4*)tgt;

    float acc = 0.0f;
    for (long long i = tid; i < n4; i += stride) {
        vf4 p = __builtin_nontemporal_load(&p4[i]);   // global_load_b128, TH=NT
        vf4 t = __builtin_nontemporal_load(&t4[i]);
        acc += elem_loss(p[0], t[0]);
        acc += elem_loss(p[1], t[1]);
        acc += elem_loss(p[2], t[2]);
        acc += elem_loss(p[3], t[3]);
    }
    if (tid == 0) {  // scalar tail (n not multiple of 4) — deterministic
        for (long long r = n4 << 2; r < n; ++r)
            acc += elem_loss(pred[r], tgt[r]);
    }

    // Full-EXEC point: wave32 WMMA reduction, then LDS across the 8 waves.
    float wsum = wave_sum_wmma(acc);
    __shared__ float lds[8];
    const int lane = threadIdx.x & 31;
    const int wv   = threadIdx.x >> 5;
    if (lane == 0) lds[wv] = wsum;
    __syncthreads();
    if (threadIdx.x == 0) {
        float s = 0.0f;
        const int nw = blockDim.x >> 5;
        for (int i = 0; i < nw; ++i) s += lds[i];
        partials[blockIdx.x] = s;
    }
}

__global__ void __launch_bounds__(256)
interval_loss_finalize(const float* __restrict__ partials,
                       float* __restrict__ out, int nparts, float inv_n) {
    float acc = 0.0f;
    for (int i = threadIdx.x; i < nparts; i += blockDim.x)
        acc += partials[i];

    float wsum = wave_sum_wmma(acc);
    __shared__ float lds[8];
    const int lane = threadIdx.x & 31;
    const int wv   = threadIdx.x >> 5;
    if (lane == 0) lds[wv] = wsum;
    __syncthreads();
    if (threadIdx.x == 0) {
        float s = 0.0f;
        const int nw = blockDim.x >> 5;
        for (int i = 0; i < nw; ++i) s += lds[i];
        out[0] = s * inv_n;
    }
}

extern "C" void kernel_launch(void* const* d_in, const int* in_sizes, int n_in,
                              void* d_out, int out_size, void* d_ws, size_t ws_size,
                              hipStream_t stream) {
    const float* pred = (const float*)d_in[0];
    const float* tgt  = (const float*)d_in[1];
    float* out        = (float*)d_out;
    float* partials   = (float*)d_ws;
    const long long n = (long long)in_sizes[0];

    // 4096 blocks x 8 waves = 32K wave32's: enough concurrency for 23.3 TB/s.
    int blocks = 4096;
    if ((size_t)blocks * sizeof(float) > ws_size)
        blocks = (int)(ws_size / sizeof(float));
    if (blocks < 1) blocks = 1;

    interval_loss_partial<<<blocks, 256, 0, stream>>>(pred, tgt, partials, n);
    interval_loss_finalize<<<1, 256, 0, stream>>>(partials, out, blocks,
                                                  1.0f / (float)n);
}